// Stochastic_encoder1_5566277616136
// MI455X (gfx1250) — compile-verified
//
#include <hip/hip_runtime.h>
#include <math.h>

typedef float v2f __attribute__((ext_vector_type(2)));
typedef float v8f __attribute__((ext_vector_type(8)));

#define IN_C  256
#define HID_C 64
#define OUT_C 32

// ---------------- degree / normalization ----------------

__global__ void k_init_deg(float* __restrict__ deg, int n) {
    int i = blockIdx.x * blockDim.x + threadIdx.x;
    if (i < n) deg[i] = 1.0f;  // self loop
}

__global__ void k_count_deg(const int* __restrict__ edges, int E, float* __restrict__ deg) {
    int e = blockIdx.x * blockDim.x + threadIdx.x;
    if (e < E) {
        int dst = edges[E + e];
        atomicAdd(&deg[dst], 1.0f);
    }
}

__global__ void k_dinv(float* __restrict__ deg, int n) {
    int i = blockIdx.x * blockDim.x + threadIdx.x;
    if (i < n) {
        float d = deg[i];
        deg[i] = (d > 0.0f) ? (1.0f / sqrtf(d)) : 0.0f;
    }
}

// ---------------- GEMM 1: hproj = X[ n x 256 ] @ W1[ 256 x 64 ]  (fp32 WMMA) --------
// One wave computes one 16x16 output tile. Block = 4 waves = the 4 N-tiles of HID_C=64.
// A layout (16x4 f32): lanes 0-15 hold K=0,1 ; lanes 16-31 hold K=2,3 ; M = lane%16.
// B layout (4x16 f32): lanes 0-15 hold K=0,1 ; lanes 16-31 hold K=2,3 ; N = lane%16.
// D layout: VGPR r -> row m0 + 8*(lane>=16) + r, col n0 + lane%16.

__global__ void __launch_bounds__(128)
k_gemm1_wmma(const float* __restrict__ X, const float* __restrict__ W,
             float* __restrict__ out, int nrows) {
    const int lane = threadIdx.x & 31;
    const int wave = threadIdx.x >> 5;          // 0..3 -> N tile
    const int m0   = blockIdx.x * 16;
    const int n0   = wave * 16;
    const int l    = lane & 15;
    const int half = lane >> 4;                 // 0 or 1

    int rowA = m0 + l;
    if (rowA > nrows - 1) rowA = nrows - 1;     // clamp (keeps EXEC all-ones)

    const float* xrow = X + (size_t)rowA * IN_C + 2 * half;
    const float* wcol = W + (size_t)(2 * half) * HID_C + (n0 + l);

    v8f c = {0.f, 0.f, 0.f, 0.f, 0.f, 0.f, 0.f, 0.f};
#pragma unroll 4
    for (int k0 = 0; k0 < IN_C; k0 += 4) {
        float2 av = *(const float2*)(xrow + k0);
        v2f a, b;
        a.x = av.x;
        a.y = av.y;
        b.x = wcol[(size_t)k0 * HID_C];
        b.y = wcol[(size_t)(k0 + 1) * HID_C];
        c = __builtin_amdgcn_wmma_f32_16x16x4_f32(false, a, false, b,
                                                  (short)0, c, false, false);
    }

#pragma unroll
    for (int i = 0; i < 8; ++i) {
        int rowD = m0 + half * 8 + i;
        if (rowD < nrows)
            out[(size_t)rowD * HID_C + (n0 + l)] = c[i];
    }
}

// ---------------- GEMM 2+3 fused: muproj = H @ Wmu, lsproj = H @ Wls  (fp32 WMMA) ----
// Block = 4 waves: waves 0,1 -> mu N-tiles 0,1 ; waves 2,3 -> ls N-tiles 0,1.

__global__ void __launch_bounds__(128)
k_gemm23_wmma(const float* __restrict__ H,
              const float* __restrict__ Wmu, const float* __restrict__ Wls,
              float* __restrict__ muproj, float* __restrict__ lsproj, int nrows) {
    const int lane = threadIdx.x & 31;
    const int wave = threadIdx.x >> 5;
    const int mat  = wave >> 1;                 // 0 = mu, 1 = ls
    const int m0   = blockIdx.x * 16;
    const int n0   = (wave & 1) * 16;
    const int l    = lane & 15;
    const int half = lane >> 4;

    const float* W  = mat ? Wls : Wmu;
    float* out      = mat ? lsproj : muproj;

    int rowA = m0 + l;
    if (rowA > nrows - 1) rowA = nrows - 1;

    const float* hrow = H + (size_t)rowA * HID_C + 2 * half;
    const float* wcol = W + (size_t)(2 * half) * OUT_C + (n0 + l);

    v8f c = {0.f, 0.f, 0.f, 0.f, 0.f, 0.f, 0.f, 0.f};
#pragma unroll
    for (int k0 = 0; k0 < HID_C; k0 += 4) {
        float2 av = *(const float2*)(hrow + k0);
        v2f a, b;
        a.x = av.x;
        a.y = av.y;
        b.x = wcol[(size_t)k0 * OUT_C];
        b.y = wcol[(size_t)(k0 + 1) * OUT_C];
        c = __builtin_amdgcn_wmma_f32_16x16x4_f32(false, a, false, b,
                                                  (short)0, c, false, false);
    }

#pragma unroll
    for (int i = 0; i < 8; ++i) {
        int rowD = m0 + half * 8 + i;
        if (rowD < nrows)
            out[(size_t)rowD * OUT_C + (n0 + l)] = c[i];
    }
}

// ---------------- layer 1: self-loop + bias init, then edge scatter -----------------

__global__ void k_self1(const float* __restrict__ hproj, const float* __restrict__ dinv,
                        const float* __restrict__ b1, float* __restrict__ h, int total) {
    int t = blockIdx.x * blockDim.x + threadIdx.x;   // total = n * HID_C
    if (t < total) {
        int i = t >> 6;        // / 64
        int f = t & 63;
        float di = dinv[i];
        h[t] = hproj[t] * di * di + b1[f];
    }
}

__global__ void k_scatter1(const int* __restrict__ edges, int E,
                           const float* __restrict__ hproj,
                           const float* __restrict__ dinv,
                           float* __restrict__ h) {
    int t = blockIdx.x * blockDim.x + threadIdx.x;   // E * 16 threads, 4 floats each
    if (t >= E * 16) return;
    int e = t >> 4;
    int q = (t & 15) << 2;
    int src = edges[e];
    int dst = edges[E + e];
    float norm = dinv[src] * dinv[dst];
    float4 v = *(const float4*)(hproj + (size_t)src * HID_C + q);
    float* o = h + (size_t)dst * HID_C + q;
    atomicAdd(o + 0, v.x * norm);
    atomicAdd(o + 1, v.y * norm);
    atomicAdd(o + 2, v.z * norm);
    atomicAdd(o + 3, v.w * norm);
}

// ---------------- layer 2/3: self-loop + bias into d_out, then edge scatter ---------

__global__ void k_self2(const float* __restrict__ muproj, const float* __restrict__ lsproj,
                        const float* __restrict__ dinv,
                        const float* __restrict__ bmu, const float* __restrict__ bls,
                        float* __restrict__ outbuf, int n) {
    int t = blockIdx.x * blockDim.x + threadIdx.x;   // n * OUT_C threads
    int total = n * OUT_C;
    if (t < total) {
        int i = t >> 5;        // / 32
        int f = t & 31;
        float di = dinv[i];
        float di2 = di * di;
        outbuf[t]         = muproj[t] * di2 + bmu[f];
        outbuf[total + t] = lsproj[t] * di2 + bls[f];
    }
}

__global__ void k_scatter2(const int* __restrict__ edges, int E,
                           const float* __restrict__ muproj,
                           const float* __restrict__ lsproj,
                           const float* __restrict__ dinv,
                           float* __restrict__ outbuf, int n) {
    int t = blockIdx.x * blockDim.x + threadIdx.x;   // E * 16 threads
    if (t >= E * 16) return;
    int e   = t >> 4;
    int r   = t & 15;
    int mat = r >> 3;          // 0 = mu, 1 = ls
    int q   = (r & 7) << 2;    // 8 chunks x 4 floats = 32
    int src = edges[e];
    int dst = edges[E + e];
    float norm = dinv[src] * dinv[dst];
    const float* proj = mat ? lsproj : muproj;
    float* outb = outbuf + (size_t)mat * ((size_t)n * OUT_C);
    float4 v = *(const float4*)(proj + (size_t)src * OUT_C + q);
    float* o = outb + (size_t)dst * OUT_C + q;
    atomicAdd(o + 0, v.x * norm);
    atomicAdd(o + 1, v.y * norm);
    atomicAdd(o + 2, v.z * norm);
    atomicAdd(o + 3, v.w * norm);
}

// ---------------- launcher ----------------

extern "C" void kernel_launch(void* const* d_in, const int* in_sizes, int n_in,
                              void* d_out, int out_size, void* d_ws, size_t ws_size,
                              hipStream_t stream) {
    const float* x   = (const float*)d_in[0];
    const int*   ei  = (const int*)d_in[1];
    const float* W1  = (const float*)d_in[2];
    const float* b1  = (const float*)d_in[3];
    const float* Wmu = (const float*)d_in[4];
    const float* bmu = (const float*)d_in[5];
    const float* Wls = (const float*)d_in[6];
    const float* bls = (const float*)d_in[7];
    float* outbuf = (float*)d_out;

    const int n = in_sizes[0] / IN_C;   // 100000
    const int E = in_sizes[1] / 2;      // 1600000

    // workspace layout
    char* ws = (char*)d_ws;
    size_t off = 0;
    auto carve = [&](size_t bytes) -> float* {
        float* p = (float*)(ws + off);
        off += (bytes + 255) & ~(size_t)255;
        return p;
    };
    float* dinv   = carve((size_t)n * sizeof(float));
    float* hproj  = carve((size_t)n * HID_C * sizeof(float));
    float* h      = carve((size_t)n * HID_C * sizeof(float));
    float* muproj = carve((size_t)n * OUT_C * sizeof(float));
    float* lsproj = carve((size_t)n * OUT_C * sizeof(float));
    (void)ws_size;

    const int B = 256;
    dim3 blk(B);

    // degrees -> dinv (in place)
    k_init_deg <<<dim3((n + B - 1) / B), blk, 0, stream>>>(dinv, n);
    k_count_deg<<<dim3((E + B - 1) / B), blk, 0, stream>>>(ei, E, dinv);
    k_dinv     <<<dim3((n + B - 1) / B), blk, 0, stream>>>(dinv, n);

    // layer 1
    int mtiles = (n + 15) / 16;
    k_gemm1_wmma<<<dim3(mtiles), dim3(128), 0, stream>>>(x, W1, hproj, n);
    k_self1   <<<dim3((n * HID_C + B - 1) / B), blk, 0, stream>>>(hproj, dinv, b1, h, n * HID_C);
    k_scatter1<<<dim3((E * 16 + B - 1) / B), blk, 0, stream>>>(ei, E, hproj, dinv, h);

    // layers 2+3 (fused)
    k_gemm23_wmma<<<dim3(mtiles), dim3(128), 0, stream>>>(h, Wmu, Wls, muproj, lsproj, n);
    k_self2   <<<dim3((n * OUT_C + B - 1) / B), blk, 0, stream>>>(muproj, lsproj, dinv, bmu, bls, outbuf, n);
    k_scatter2<<<dim3((E * 16 + B - 1) / B), blk, 0, stream>>>(ei, E, muproj, lsproj, dinv, outbuf, n);
}